// Memory_60189671686643
// MI455X (gfx1250) — compile-verified
//
#include <hip/hip_runtime.h>
#include <hip/hip_bf16.h>

typedef float v2f __attribute__((ext_vector_type(2)));
typedef float v8f __attribute__((ext_vector_type(8)));

#define DIMC 512
#define HEADC 8
#define SLOTC 112
#define RADIUSF 16.0f
#define TE_ELEMS 8388608   // 16384*512
#define LDX 520            // padded row stride for 16x512 tile
#define LDP 120            // padded row stride for 16x112 tile

__device__ __forceinline__ v8f wmma4(v2f a, v2f b, v8f c) {
    // D(16x16,f32) = A(16x4,f32) * B(4x16,f32) + C
    return __builtin_amdgcn_wmma_f32_16x16x4_f32(false, a, false, b, (short)0, c, false, false);
}

// ---------------- precompute kernels ----------------

__global__ void k_zero_tail(float* out) {
    int i = threadIdx.x;
    if (i < 33) out[(size_t)TE_ELEMS * 2 + i] = 0.0f;
}

__global__ void k_rownorm(const float* __restrict__ key_mem,
                          const float* __restrict__ value_mem,
                          float* __restrict__ KN, float* __restrict__ VN) {
    int r = blockIdx.x * blockDim.x + threadIdx.x;
    if (r < 896) {                       // key_mem rows: (HEAD*SLOT) x 64
        const float* in = key_mem + r * 64;
        float s = 0.f;
        for (int d = 0; d < 64; d++) s += in[d] * in[d];
        float inv = 1.0f / fmaxf(sqrtf(s), 1e-12f);
        for (int d = 0; d < 64; d++) KN[r * 64 + d] = in[d] * inv;
    } else if (r < 1008) {               // value_mem rows: 112 x 512
        int rr = r - 896;
        const float* in = value_mem + rr * DIMC;
        float s = 0.f;
        for (int d = 0; d < DIMC; d++) s += in[d] * in[d];
        float inv = 1.0f / fmaxf(sqrtf(s), 1e-12f);
        for (int d = 0; d < DIMC; d++) VN[rr * DIMC + d] = in[d] * inv;
    }
}

__global__ void k_transpose_vm(const float* __restrict__ vm, float* __restrict__ VMT) {
    int idx = blockIdx.x * blockDim.x + threadIdx.x;
    if (idx < DIMC * SLOTC) {
        int d = idx / SLOTC, s = idx % SLOTC;
        VMT[idx] = vm[s * DIMC + d];     // VMT[d][s] = value_mem[s][d]
    }
}

// VWT[h][d][s] = sum_c value_mem[s][c] * Wl[d][h*512 + c]
__global__ void k_vwt(const float* __restrict__ vm, const float* __restrict__ Wl,
                      float* __restrict__ VWT) {
    int o = blockIdx.x * blockDim.x + threadIdx.x;
    if (o >= HEADC * DIMC * SLOTC) return;
    int h = o / (DIMC * SLOTC);
    int rem = o % (DIMC * SLOTC);
    int d = rem / SLOTC, s = rem % SLOTC;
    const float* a = vm + (size_t)s * DIMC;
    const float* w = Wl + (size_t)d * (DIMC * HEADC) + h * DIMC;
    float acc = 0.f;
    for (int c = 0; c < DIMC; c++) acc += a[c] * w[c];
    VWT[o] = acc;
}

__global__ void k_contrastive(const float* __restrict__ VN, float* __restrict__ out) {
    __shared__ float red[256];
    float local = 0.f;
    for (int idx = threadIdx.x; idx < SLOTC * SLOTC; idx += 256) {
        int i = idx / SLOTC, j = idx % SLOTC;
        const float* a = VN + (size_t)i * DIMC;
        const float* b = VN + (size_t)j * DIMC;
        float dot = 0.f;
        for (int c = 0; c < DIMC; c++) dot += a[c] * b[c];
        local += fabsf(((i == j) ? 1.0f : 0.0f) - dot);
    }
    red[threadIdx.x] = local;
    __syncthreads();
    for (int s = 128; s > 0; s >>= 1) {
        if (threadIdx.x < s) red[threadIdx.x] += red[threadIdx.x + s];
        __syncthreads();
    }
    if (threadIdx.x == 0) out[(size_t)TE_ELEMS * 2 + 32] = 0.5f * red[0];
}

// ---------------- read path: te_fusion ----------------
__global__ __launch_bounds__(256)
void k_read_path(const float* __restrict__ q, const float* __restrict__ Wq,
                 const float* __restrict__ bq, const float* __restrict__ bl,
                 const float* __restrict__ g1, const float* __restrict__ b1,
                 const float* __restrict__ g2, const float* __restrict__ b2,
                 const float* __restrict__ KN, const float* __restrict__ VWT,
                 float* __restrict__ te) {
    __shared__ float sX[16 * LDX];
    __shared__ float sP[16 * LDP];
    __shared__ float sMu[16], sRs[16];

    const int tid  = threadIdx.x;
    const int lane = tid & 31;
    const int wave = tid >> 5;
    const int hlf  = lane >> 4;      // 0: lanes 0-15, 1: lanes 16-31
    const int l15  = lane & 15;
    const int tok0 = blockIdx.x * 16;
    const int nb   = wave * 64;      // this wave's N-column base

    // ---- GEMM1: eq = q @ Wq^T + bq  (16 x 512) ----
    {
        v8f a0 = {}, a1 = {}, a2 = {}, a3 = {};
        const float* Arow = q + (size_t)(tok0 + l15) * DIMC;
        for (int k0 = 0; k0 < DIMC; k0 += 4) {
            v2f av = *(const v2f*)(Arow + k0 + 2 * hlf);
            v2f bv0 = *(const v2f*)(Wq + (size_t)(nb +  0 + l15) * DIMC + k0 + 2 * hlf);
            v2f bv1 = *(const v2f*)(Wq + (size_t)(nb + 16 + l15) * DIMC + k0 + 2 * hlf);
            v2f bv2 = *(const v2f*)(Wq + (size_t)(nb + 32 + l15) * DIMC + k0 + 2 * hlf);
            v2f bv3 = *(const v2f*)(Wq + (size_t)(nb + 48 + l15) * DIMC + k0 + 2 * hlf);
            a0 = wmma4(av, bv0, a0);
            a1 = wmma4(av, bv1, a1);
            a2 = wmma4(av, bv2, a2);
            a3 = wmma4(av, bv3, a3);
        }
#pragma unroll
        for (int v = 0; v < 8; v++) {
            int m = v + 8 * hlf;
            sX[m * LDX + nb +  0 + l15] = a0[v] + bq[nb +  0 + l15];
            sX[m * LDX + nb + 16 + l15] = a1[v] + bq[nb + 16 + l15];
            sX[m * LDX + nb + 32 + l15] = a2[v] + bq[nb + 32 + l15];
            sX[m * LDX + nb + 48 + l15] = a3[v] + bq[nb + 48 + l15];
        }
    }
    __syncthreads();

    // ---- per-(token,head) l2 normalize (64-dim chunks) ----
    if (tid < 128) {
        int t = tid >> 3, h = tid & 7;
        int base = t * LDX + h * 64;
        float s = 0.f;
        for (int d = 0; d < 64; d++) { float x = sX[base + d]; s += x * x; }
        float inv = 1.0f / fmaxf(sqrtf(s), 1e-12f);
        for (int d = 0; d < 64; d++) sX[base + d] *= inv;
    }
    __syncthreads();

    // ---- per-head: sim -> softmax -> vir accumulate ----
    v8f c0 = {}, c1 = {}, c2 = {}, c3 = {};
    for (int h = 0; h < HEADC; h++) {
        if (wave < 7) {   // 7 waves x 16 slots = 112
            v8f acc = {};
            const int srow = h * SLOTC + wave * 16 + l15;
            for (int k0 = 0; k0 < 64; k0 += 4) {
                v2f av = *(const v2f*)(&sX[l15 * LDX + h * 64 + k0 + 2 * hlf]);
                v2f bv = *(const v2f*)(KN + (size_t)srow * 64 + k0 + 2 * hlf);
                acc = wmma4(av, bv, acc);
            }
#pragma unroll
            for (int v = 0; v < 8; v++)
                sP[(v + 8 * hlf) * LDP + wave * 16 + l15] = RADIUSF * acc[v];
        }
        __syncthreads();
        if (tid < 16) {   // softmax over 112 slots, one token per thread
            float mx = -1e30f;
            for (int s = 0; s < SLOTC; s++) mx = fmaxf(mx, sP[tid * LDP + s]);
            float sum = 0.f;
            for (int s = 0; s < SLOTC; s++) {
                float e = __expf(sP[tid * LDP + s] - mx);
                sP[tid * LDP + s] = e; sum += e;
            }
            float inv = 1.0f / sum;
            for (int s = 0; s < SLOTC; s++) sP[tid * LDP + s] *= inv;
        }
        __syncthreads();
        {   // vir[t,d] += sum_s p[t,s] * VWT[h][d][s]
            const float* Bb = VWT + (size_t)h * DIMC * SLOTC;
            for (int k0 = 0; k0 < SLOTC; k0 += 4) {
                v2f av = *(const v2f*)(&sP[l15 * LDP + k0 + 2 * hlf]);
                v2f bv0 = *(const v2f*)(Bb + (size_t)(nb +  0 + l15) * SLOTC + k0 + 2 * hlf);
                v2f bv1 = *(const v2f*)(Bb + (size_t)(nb + 16 + l15) * SLOTC + k0 + 2 * hlf);
                v2f bv2 = *(const v2f*)(Bb + (size_t)(nb + 32 + l15) * SLOTC + k0 + 2 * hlf);
                v2f bv3 = *(const v2f*)(Bb + (size_t)(nb + 48 + l15) * SLOTC + k0 + 2 * hlf);
                c0 = wmma4(av, bv0, c0);
                c1 = wmma4(av, bv1, c1);
                c2 = wmma4(av, bv2, c2);
                c3 = wmma4(av, bv3, c3);
            }
        }
        __syncthreads();
    }

    // ---- vir raw (+bl) into sX ----
#pragma unroll
    for (int v = 0; v < 8; v++) {
        int m = v + 8 * hlf;
        sX[m * LDX + nb +  0 + l15] = c0[v] + bl[nb +  0 + l15];
        sX[m * LDX + nb + 16 + l15] = c1[v] + bl[nb + 16 + l15];
        sX[m * LDX + nb + 32 + l15] = c2[v] + bl[nb + 32 + l15];
        sX[m * LDX + nb + 48 + l15] = c3[v] + bl[nb + 48 + l15];
    }
    __syncthreads();

    // ---- LN2 stats ----
    if (tid < 16) {
        float s = 0.f, s2 = 0.f;
        for (int j = 0; j < DIMC; j++) { float x = sX[tid * LDX + j]; s += x; s2 += x * x; }
        float mu = s / DIMC;
        sMu[tid] = mu; sRs[tid] = rsqrtf(s2 / DIMC - mu * mu + 1e-5f);
    }
    __syncthreads();
    // x = q + LN2(vir)
    for (int idx = tid; idx < 16 * DIMC; idx += 256) {
        int t = idx >> 9, j = idx & 511;
        float y = (sX[t * LDX + j] - sMu[t]) * sRs[t] * g2[j] + b2[j];
        sX[t * LDX + j] = y + q[(size_t)(tok0 + t) * DIMC + j];
    }
    __syncthreads();
    if (tid < 16) {
        float s = 0.f, s2 = 0.f;
        for (int j = 0; j < DIMC; j++) { float x = sX[tid * LDX + j]; s += x; s2 += x * x; }
        float mu = s / DIMC;
        sMu[tid] = mu; sRs[tid] = rsqrtf(s2 / DIMC - mu * mu + 1e-5f);
    }
    __syncthreads();
    for (int idx = tid; idx < 16 * DIMC; idx += 256) {
        int t = idx >> 9, j = idx & 511;
        te[(size_t)(tok0 + t) * DIMC + j] =
            (sX[t * LDX + j] - sMu[t]) * sRs[t] * g1[j] + b1[j];
    }
}

// ---------------- write path: tr_fusion + recon ----------------
__global__ __launch_bounds__(256)
void k_write_path(const float* __restrict__ q, const float* __restrict__ val,
                  const float* __restrict__ Wv, const float* __restrict__ bv,
                  const float* __restrict__ g1, const float* __restrict__ b1,
                  const float* __restrict__ g3, const float* __restrict__ b3,
                  const float* __restrict__ VN, const float* __restrict__ VMT,
                  float* __restrict__ tr, float* __restrict__ recon) {
    __shared__ float sX[16 * LDX];
    __shared__ float sP[16 * LDP];
    __shared__ float sMu[16], sRs[16];

    const int tid  = threadIdx.x;
    const int lane = tid & 31;
    const int wave = tid >> 5;
    const int hlf  = lane >> 4;
    const int l15  = lane & 15;
    const int tok0 = blockIdx.x * 16;
    const int nb   = wave * 64;

    // ---- ev = value @ Wv^T + bv ----
    {
        v8f a0 = {}, a1 = {}, a2 = {}, a3 = {};
        const float* Arow = val + (size_t)(tok0 + l15) * DIMC;
        for (int k0 = 0; k0 < DIMC; k0 += 4) {
            v2f av = *(const v2f*)(Arow + k0 + 2 * hlf);
            v2f bv0 = *(const v2f*)(Wv + (size_t)(nb +  0 + l15) * DIMC + k0 + 2 * hlf);
            v2f bv1 = *(const v2f*)(Wv + (size_t)(nb + 16 + l15) * DIMC + k0 + 2 * hlf);
            v2f bv2 = *(const v2f*)(Wv + (size_t)(nb + 32 + l15) * DIMC + k0 + 2 * hlf);
            v2f bv3 = *(const v2f*)(Wv + (size_t)(nb + 48 + l15) * DIMC + k0 + 2 * hlf);
            a0 = wmma4(av, bv0, a0);
            a1 = wmma4(av, bv1, a1);
            a2 = wmma4(av, bv2, a2);
            a3 = wmma4(av, bv3, a3);
        }
#pragma unroll
        for (int v = 0; v < 8; v++) {
            int m = v + 8 * hlf;
            sX[m * LDX + nb +  0 + l15] = a0[v] + bv[nb +  0 + l15];
            sX[m * LDX + nb + 16 + l15] = a1[v] + bv[nb + 16 + l15];
            sX[m * LDX + nb + 32 + l15] = a2[v] + bv[nb + 32 + l15];
            sX[m * LDX + nb + 48 + l15] = a3[v] + bv[nb + 48 + l15];
        }
    }
    __syncthreads();

    // ---- l2 normalize ev rows (512-dim) ----
    if (tid < 16) {
        float s = 0.f;
        for (int j = 0; j < DIMC; j++) { float x = sX[tid * LDX + j]; s += x * x; }
        float inv = 1.0f / fmaxf(sqrtf(s), 1e-12f);
        for (int j = 0; j < DIMC; j++) sX[tid * LDX + j] *= inv;
    }
    __syncthreads();

    // ---- value_sim = evn @ VN^T  (16 x 112, K=512) ----
    if (wave < 7) {
        v8f acc = {};
        const int srow = wave * 16 + l15;
        for (int k0 = 0; k0 < DIMC; k0 += 4) {
            v2f av = *(const v2f*)(&sX[l15 * LDX + k0 + 2 * hlf]);
            v2f bvv = *(const v2f*)(VN + (size_t)srow * DIMC + k0 + 2 * hlf);
            acc = wmma4(av, bvv, acc);
        }
#pragma unroll
        for (int v = 0; v < 8; v++)
            sP[(v + 8 * hlf) * LDP + wave * 16 + l15] = RADIUSF * acc[v];
    }
    __syncthreads();
    if (tid < 16) {
        float mx = -1e30f;
        for (int s = 0; s < SLOTC; s++) mx = fmaxf(mx, sP[tid * LDP + s]);
        float sum = 0.f;
        for (int s = 0; s < SLOTC; s++) {
            float e = __expf(sP[tid * LDP + s] - mx);
            sP[tid * LDP + s] = e; sum += e;
        }
        float inv = 1.0f / sum;
        for (int s = 0; s < SLOTC; s++) sP[tid * LDP + s] *= inv;
    }
    __syncthreads();

    // ---- aud = p @ value_mem  (16 x 512, K=112) via VMT[d][s] ----
    v8f c0 = {}, c1 = {}, c2 = {}, c3 = {};
    for (int k0 = 0; k0 < SLOTC; k0 += 4) {
        v2f av = *(const v2f*)(&sP[l15 * LDP + k0 + 2 * hlf]);
        v2f bv0 = *(const v2f*)(VMT + (size_t)(nb +  0 + l15) * SLOTC + k0 + 2 * hlf);
        v2f bv1 = *(const v2f*)(VMT + (size_t)(nb + 16 + l15) * SLOTC + k0 + 2 * hlf);
        v2f bv2 = *(const v2f*)(VMT + (size_t)(nb + 32 + l15) * SLOTC + k0 + 2 * hlf);
        v2f bv3 = *(const v2f*)(VMT + (size_t)(nb + 48 + l15) * SLOTC + k0 + 2 * hlf);
        c0 = wmma4(av, bv0, c0);
        c1 = wmma4(av, bv1, c1);
        c2 = wmma4(av, bv2, c2);
        c3 = wmma4(av, bv3, c3);
    }
#pragma unroll
    for (int v = 0; v < 8; v++) {
        int m = v + 8 * hlf;
        sX[m * LDX + nb +  0 + l15] = c0[v];
        sX[m * LDX + nb + 16 + l15] = c1[v];
        sX[m * LDX + nb + 32 + l15] = c2[v];
        sX[m * LDX + nb + 48 + l15] = c3[v];
    }
    __syncthreads();

    // ---- cos/recon + LN3 stats ----
    if (tid < 16) {
        const float* vrow = val + (size_t)(tok0 + tid) * DIMC;
        float dot = 0.f, na = 0.f, nv = 0.f, s = 0.f;
        for (int j = 0; j < DIMC; j++) {
            float a = sX[tid * LDX + j];
            float b = vrow[j];
            dot += a * b; na += a * a; nv += b * b; s += a;
        }
        float cosv = dot / fmaxf(sqrtf(na) * sqrtf(nv), 1e-8f);
        atomicAdd(&recon[(tok0 + tid) >> 9], fabsf(1.0f - cosv));
        float mu = s / DIMC;
        sMu[tid] = mu; sRs[tid] = rsqrtf(na / DIMC - mu * mu + 1e-5f);
    }
    __syncthreads();
    // x = q + LN3(aud)
    for (int idx = tid; idx < 16 * DIMC; idx += 256) {
        int t = idx >> 9, j = idx & 511;
        float y = (sX[t * LDX + j] - sMu[t]) * sRs[t] * g3[j] + b3[j];
        sX[t * LDX + j] = y + q[(size_t)(tok0 + t) * DIMC + j];
    }
    __syncthreads();
    if (tid < 16) {
        float s = 0.f, s2 = 0.f;
        for (int j = 0; j < DIMC; j++) { float x = sX[tid * LDX + j]; s += x; s2 += x * x; }
        float mu = s / DIMC;
        sMu[tid] = mu; sRs[tid] = rsqrtf(s2 / DIMC - mu * mu + 1e-5f);
    }
    __syncthreads();
    for (int idx = tid; idx < 16 * DIMC; idx += 256) {
        int t = idx >> 9, j = idx & 511;
        tr[(size_t)(tok0 + t) * DIMC + j] =
            (sX[t * LDX + j] - sMu[t]) * sRs[t] * g1[j] + b1[j];
    }
}

// ---------------- launcher ----------------
extern "C" void kernel_launch(void* const* d_in, const int* in_sizes, int n_in,
                              void* d_out, int out_size, void* d_ws, size_t ws_size,
                              hipStream_t stream) {
    const float* q         = (const float*)d_in[0];
    const float* val       = (const float*)d_in[1];
    const float* key_mem   = (const float*)d_in[2];
    const float* value_mem = (const float*)d_in[3];
    const float* Wq = (const float*)d_in[4];
    const float* bq = (const float*)d_in[5];
    const float* Wv = (const float*)d_in[6];
    const float* bv = (const float*)d_in[7];
    const float* Wl = (const float*)d_in[8];
    const float* bl = (const float*)d_in[9];
    const float* g1 = (const float*)d_in[10];
    const float* b1 = (const float*)d_in[11];
    const float* g2 = (const float*)d_in[12];
    const float* b2 = (const float*)d_in[13];
    const float* g3 = (const float*)d_in[14];
    const float* b3 = (const float*)d_in[15];

    float* out = (float*)d_out;
    float* ws  = (float*)d_ws;
    float* KN  = ws;                 // 896*64   = 57344
    float* VN  = ws + 57344;         // 112*512  = 57344
    float* VMT = ws + 114688;        // 512*112  = 57344
    float* VWT = ws + 172032;        // 8*512*112 = 458752
    float* te    = out;
    float* tr    = out + (size_t)TE_ELEMS;
    float* recon = out + (size_t)TE_ELEMS * 2;

    hipLaunchKernelGGL(k_zero_tail, dim3(1), dim3(64), 0, stream, out);
    hipLaunchKernelGGL(k_rownorm, dim3(8), dim3(128), 0, stream, key_mem, value_mem, KN, VN);
    hipLaunchKernelGGL(k_transpose_vm, dim3((DIMC * SLOTC + 255) / 256), dim3(256), 0, stream,
                       value_mem, VMT);
    hipLaunchKernelGGL(k_vwt, dim3((HEADC * DIMC * SLOTC + 255) / 256), dim3(256), 0, stream,
                       value_mem, Wl, VWT);
    hipLaunchKernelGGL(k_contrastive, dim3(1), dim3(256), 0, stream, VN, out);
    hipLaunchKernelGGL(k_read_path, dim3(1024), dim3(256), 0, stream,
                       q, Wq, bq, bl, g1, b1, g2, b2, KN, VWT, te);
    hipLaunchKernelGGL(k_write_path, dim3(1024), dim3(256), 0, stream,
                       q, val, Wv, bv, g1, b1, g3, b3, VN, VMT, tr, recon);
}